// ToyLLM_29326036697397
// MI455X (gfx1250) — compile-verified
//
#include <hip/hip_runtime.h>
#include <cstdint>

typedef __attribute__((ext_vector_type(2))) float v2f;
typedef __attribute__((ext_vector_type(8))) float v8f;

#define VOCAB 32000
#define HDIM  128
#define BATCH 16
#define SEQ   256
#define H3    384

__device__ __forceinline__ v8f wmma_f32_16x16x4(v2f a, v2f b, v8f c) {
  // 8 args: (neg_a, A, neg_b, B, c_mod, C, reuse_a, reuse_b)
  return __builtin_amdgcn_wmma_f32_16x16x4_f32(false, a, false, b, (short)0, c,
                                               false, false);
}

__device__ __forceinline__ float sigmoidf_(float x) {
  return 1.0f / (1.0f + __expf(-x));
}

// ---------------------------------------------------------------------------
// Kernel 1: xi[m,0:384] = embed[x[m]] @ Wi + bi, m in [0, B*T)
// grid.x = 256 M-tiles; 128 threads = 4 waves; each wave: 6 N-tiles (3 pairs)
// ---------------------------------------------------------------------------
__global__ __launch_bounds__(128)
void k_embed_xi(const int* __restrict__ x, const float* __restrict__ embed,
                const float* __restrict__ Wi, const float* __restrict__ bi,
                float* __restrict__ xi) {
  const int lane = threadIdx.x & 31;
  const int wave = threadIdx.x >> 5;
  const int half = lane >> 4;   // K sub-pair select
  const int lr   = lane & 15;   // row (A) / col (B,C)
  const int m0   = blockIdx.x * 16;

  const int tok = x[m0 + lr];
  const float* arow = embed + (size_t)tok * HDIM;
  v2f a[32];
  #pragma unroll
  for (int i = 0; i < 32; ++i)
    a[i] = *(const v2f*)(arow + 4 * i + 2 * half);

  #pragma unroll
  for (int p = 0; p < 3; ++p) {
    const int nt0 = wave * 6 + p * 2;
    const int n0  = nt0 * 16 + lr;
    const int n1  = n0 + 16;
    v8f acc0 = {};
    v8f acc1 = {};
    #pragma unroll
    for (int i = 0; i < 32; ++i) {
      const int k = 4 * i + 2 * half;
      v2f b0, b1;
      b0.x = Wi[(size_t)k * H3 + n0];
      b0.y = Wi[(size_t)(k + 1) * H3 + n0];
      b1.x = Wi[(size_t)k * H3 + n1];
      b1.y = Wi[(size_t)(k + 1) * H3 + n1];
      acc0 = wmma_f32_16x16x4(a[i], b0, acc0);
      acc1 = wmma_f32_16x16x4(a[i], b1, acc1);
    }
    const float bias0 = bi[n0], bias1 = bi[n1];
    #pragma unroll
    for (int v = 0; v < 8; ++v) {
      const int m = m0 + v + 8 * half;
      xi[(size_t)m * H3 + n0] = acc0[v] + bias0;
      xi[(size_t)m * H3 + n1] = acc1[v] + bias1;
    }
  }
}

// ---------------------------------------------------------------------------
// Kernel 2: sequential GRU scan. Single workgroup, 768 threads = 24 waves,
// each wave owns one 16-wide N-tile of hh = h @ Wh (M = batch = 16).
// h (8 KB) and hh (24 KB) live in LDS; Wh streams from L2.
// ---------------------------------------------------------------------------
__global__ __launch_bounds__(768)
void k_gru_scan(const float* __restrict__ h0, const float* __restrict__ Wh,
                const float* __restrict__ bhn, const float* __restrict__ xi,
                float* __restrict__ hs, float* __restrict__ carry) {
  __shared__ float h_lds[BATCH * HDIM];
  __shared__ float hh_lds[BATCH * H3];
  const int tid  = threadIdx.x;
  const int lane = tid & 31;
  const int wave = tid >> 5;       // 0..23 -> N tile
  const int half = lane >> 4;
  const int lr   = lane & 15;
  const int n    = wave * 16 + lr;

  for (int idx = tid; idx < BATCH * HDIM; idx += 768) h_lds[idx] = h0[idx];
  __syncthreads();

  for (int t = 0; t < SEQ; ++t) {
    v8f acc = {};
    #pragma unroll
    for (int i = 0; i < 32; ++i) {
      const int k = 4 * i + 2 * half;
      v2f a = *(const v2f*)(&h_lds[lr * HDIM + k]);
      v2f b;
      b.x = Wh[(size_t)k * H3 + n];
      b.y = Wh[(size_t)(k + 1) * H3 + n];
      acc = wmma_f32_16x16x4(a, b, acc);
    }
    #pragma unroll
    for (int v = 0; v < 8; ++v) hh_lds[(v + 8 * half) * H3 + n] = acc[v];
    __syncthreads();

    for (int idx = tid; idx < BATCH * HDIM; idx += 768) {
      const int b = idx >> 7, j = idx & 127;
      const size_t m = (size_t)b * SEQ + t;
      const float ir  = xi[m * H3 + j];
      const float iz  = xi[m * H3 + 128 + j];
      const float in_ = xi[m * H3 + 256 + j];
      const float hr  = hh_lds[b * H3 + j];
      const float hz  = hh_lds[b * H3 + 128 + j];
      const float hn  = hh_lds[b * H3 + 256 + j];
      const float hprev = h_lds[idx];
      const float r  = sigmoidf_(ir + hr);
      const float z  = sigmoidf_(iz + hz);
      const float nn = tanhf(in_ + r * (hn + bhn[j]));
      const float hnew = (1.0f - z) * nn + z * hprev;
      h_lds[idx] = hnew;                 // own element only; no cross-thread race
      hs[m * HDIM + j] = hnew;
      if (t == SEQ - 1) carry[idx] = hnew;
    }
    __syncthreads();
  }
}

// ---------------------------------------------------------------------------
// Kernel 3 (dominant, store-bound): logits = hs @ Wout + bout.
// grid = (256 M-tiles, 125 N-groups); 4 waves/block, 4 N-tiles/wave (2 pairs).
// A-block held in 64 VGPRs per wave; logits stored nontemporally.
// ---------------------------------------------------------------------------
__global__ __launch_bounds__(128)
void k_logits(const float* __restrict__ hs, const float* __restrict__ Wout,
              const float* __restrict__ bout, float* __restrict__ out) {
  const int lane = threadIdx.x & 31;
  const int wave = threadIdx.x >> 5;
  const int half = lane >> 4;
  const int lr   = lane & 15;
  const int m0   = blockIdx.x * 16;

  const float* arow = hs + (size_t)(m0 + lr) * HDIM;
  v2f a[32];
  #pragma unroll
  for (int i = 0; i < 32; ++i)
    a[i] = *(const v2f*)(arow + 4 * i + 2 * half);

  #pragma unroll
  for (int p = 0; p < 2; ++p) {
    const int nt0 = (blockIdx.y * 4 + wave) * 4 + p * 2;
    const int n0  = nt0 * 16 + lr;
    const int n1  = n0 + 16;
    v8f acc0 = {};
    v8f acc1 = {};
    #pragma unroll
    for (int i = 0; i < 32; ++i) {
      const int k = 4 * i + 2 * half;
      v2f b0, b1;
      b0.x = Wout[(size_t)k * VOCAB + n0];
      b0.y = Wout[(size_t)(k + 1) * VOCAB + n0];
      b1.x = Wout[(size_t)k * VOCAB + n1];
      b1.y = Wout[(size_t)(k + 1) * VOCAB + n1];
      acc0 = wmma_f32_16x16x4(a[i], b0, acc0);
      acc1 = wmma_f32_16x16x4(a[i], b1, acc1);
    }
    const float bias0 = bout[n0], bias1 = bout[n1];
    #pragma unroll
    for (int v = 0; v < 8; ++v) {
      const int m = m0 + v + 8 * half;
      __builtin_nontemporal_store(acc0[v] + bias0, &out[(size_t)m * VOCAB + n0]);
      __builtin_nontemporal_store(acc1[v] + bias1, &out[(size_t)m * VOCAB + n1]);
    }
  }
}

extern "C" void kernel_launch(void* const* d_in, const int* in_sizes, int n_in,
                              void* d_out, int out_size, void* d_ws, size_t ws_size,
                              hipStream_t stream) {
  const int*   x     = (const int*)d_in[0];
  const float* h0    = (const float*)d_in[1];
  const float* embed = (const float*)d_in[2];
  const float* Wi    = (const float*)d_in[3];
  const float* bi    = (const float*)d_in[4];
  const float* Wh    = (const float*)d_in[5];
  const float* bhn   = (const float*)d_in[6];
  const float* Wout  = (const float*)d_in[7];
  const float* bout  = (const float*)d_in[8];
  float* out = (float*)d_out;

  float* xi_ws = (float*)d_ws;                         // [4096, 384]
  float* hs_ws = xi_ws + (size_t)BATCH * SEQ * H3;     // [4096, 128]
  float* carry = out + (size_t)BATCH * SEQ * VOCAB;    // [16, 128] tail of d_out

  k_embed_xi<<<dim3(256), dim3(128), 0, stream>>>(x, embed, Wi, bi, xi_ws);
  k_gru_scan<<<dim3(1), dim3(768), 0, stream>>>(h0, Wh, bhn, xi_ws, hs_ws, carry);
  k_logits<<<dim3(256, 125), dim3(128), 0, stream>>>(hs_ws, Wout, bout, out);
}